// CRF_zqx_42064909697650
// MI455X (gfx1250) — compile-verified
//
#include <hip/hip_runtime.h>
#include <hip/hip_bf16.h>
#include <stdint.h>

#define TAG   128
#define SEQ   1024
#define BATCH 128
#define BG    16            // batch rows per scan block
#define NBLK  (BATCH / BG)  // 8 scan blocks
#define PADC  (TAG + 4)     // padded LDS row stride (132 floats) to spread banks

typedef __attribute__((ext_vector_type(2))) float        v2f;
typedef __attribute__((ext_vector_type(8))) float        v8f;
typedef __attribute__((ext_vector_type(4))) unsigned int v4u;
typedef __attribute__((ext_vector_type(4))) int          v4i;
typedef __attribute__((ext_vector_type(8))) int          v8i;

// ---------------------------------------------------------------------------
// TDM: async 2D tile load (16 rows x 128 f32, row stride SEQ*TAG elements)
// D# group0: count=1 | lds_addr | global_addr | type=2
// D# group1: data_size=4B, tensor dims huge (no OOB), tile 128x16, dim0 stride
// ---------------------------------------------------------------------------
__device__ __forceinline__ void tdm_load_tile(const float* gsrc, uint32_t lds_off) {
  uint64_t ga = (uint64_t)(uintptr_t)gsrc;
  v4u g0;
  g0.x = 1u;                                              // count=1, user mode
  g0.y = lds_off;                                         // lds_addr (bytes)
  g0.z = (uint32_t)(ga & 0xFFFFFFFFull);                  // global_addr[31:0]
  g0.w = (uint32_t)((ga >> 32) & 0x01FFFFFFull) | (2u << 30); // [56:32] | type=2
  v8i g1;
  g1[0] = (int)(2u << 16);                 // wg_mask=0, data_size=2 (4 bytes)
  g1[1] = (int)0x7FFF0000u;                // tensor_dim0 lo16 (huge)
  g1[2] = (int)0xFFFF7FFFu;                // tensor_dim0 hi16, tensor_dim1 lo16
  g1[3] = (int)((128u << 16) | 0x7FFFu);   // tensor_dim1 hi16, tile_dim0=128
  g1[4] = (int)16;                         // tile_dim1=16, tile_dim2=0
  g1[5] = (int)(SEQ * TAG);                // tensor_dim0_stride[31:0] = 131072
  g1[6] = 0;                               // stride hi / dim1 stride lo
  g1[7] = 0;
  v4i z4 = {0, 0, 0, 0};
#if __has_include(<hip/amd_detail/amd_gfx1250_TDM.h>)
  v8i z8 = {0, 0, 0, 0, 0, 0, 0, 0};
  __builtin_amdgcn_tensor_load_to_lds(g0, g1, z4, z4, z8, 0);
#else
  __builtin_amdgcn_tensor_load_to_lds(g0, g1, z4, z4, 0);
#endif
}

// ---------------------------------------------------------------------------
// Kernel 0: zero the scalar output (harness poisons it).
// ---------------------------------------------------------------------------
__global__ void crf_zero_kernel(float* out) { out[0] = 0.0f; }

// ---------------------------------------------------------------------------
// Kernel 1: path score per batch: word2tag + tag2tag.
// ---------------------------------------------------------------------------
__global__ __launch_bounds__(256) void crf_score_kernel(
    const float* __restrict__ yp, const int* __restrict__ yt,
    const float* __restrict__ mk, const float* __restrict__ A,
    float* __restrict__ score)
{
  const int b = blockIdx.x, tid = threadIdx.x;
  float acc = 0.0f;
  for (int s = tid; s < SEQ; s += 256) {
    int   t = yt[b * SEQ + s] & (TAG - 1);
    float m = mk[b * SEQ + s];
    acc += m * yp[((size_t)b * SEQ + s) * TAG + t];
    if (s + 1 < SEQ) {
      int   t2 = yt[b * SEQ + s + 1] & (TAG - 1);
      float m2 = mk[b * SEQ + s + 1];
      acc += m * m2 * A[t * TAG + t2];
    }
  }
#pragma unroll
  for (int off = 16; off > 0; off >>= 1) acc += __shfl_xor(acc, off, 32);
  __shared__ float red[8];
  if ((tid & 31) == 0) red[tid >> 5] = acc;
  __syncthreads();
  if (tid == 0) {
    float t = 0.0f;
    for (int i = 0; i < 8; ++i) t += red[i];
    score[b] = t;
  }
}

// ---------------------------------------------------------------------------
// Kernel 2: forward scan. One block = 16 batch rows, 8 waves (256 threads).
// Per step: P = exp(PRE - rowmax); V = P x exp(A) via chained
// v_wmma_f32_16x16x4_f32 (B fragments register-resident); PRE' = mask-blend of
// rowmax + log(V) + y_pred[s]. y_pred tiles stream in via TDM (double buffer);
// the 16 mask values per step are staged to LDS (no global loads in epilogue).
// ---------------------------------------------------------------------------
__global__ __launch_bounds__(256) void crf_scan_kernel(
    const float* __restrict__ yp, const float* __restrict__ mk,
    const float* __restrict__ A, const float* __restrict__ score,
    float* __restrict__ out)
{
  __shared__ float PREsh[BG][PADC];
  __shared__ float Psh[BG][PADC];
  __shared__ float Ybuf[2][BG][TAG];
  __shared__ float Mrow[BG];
  __shared__ float Msk[BG];

  const int tid    = threadIdx.x;
  const int lane   = tid & 31;
  const int wave   = tid >> 5;
  const int b0     = blockIdx.x * BG;
  const int l16    = lane & 15;   // A-frag row M / B-frag,D col N
  const int kh     = lane >> 4;   // K-half select within 4-wide chunk
  const int u0     = wave * 16;   // this wave's output tag tile

  // B fragments: exp(A) for this wave's u-tile, all 32 K-chunks, in registers.
  v2f bf[32];
#pragma unroll
  for (int k = 0; k < 32; ++k) {
    int t0 = 4 * k + 2 * kh;
    bf[k].x = __expf(A[t0 * TAG + u0 + l16]);
    bf[k].y = __expf(A[(t0 + 1) * TAG + u0 + l16]);
  }

  // PRE = y_pred[:, 0, :]
#pragma unroll
  for (int k = 0; k < 8; ++k) {
    int idx = tid + 256 * k;
    int rr = idx >> 7, cc = idx & 127;
    PREsh[rr][cc] = yp[((size_t)(b0 + rr) * SEQ) * TAG + cc];
  }
  __syncthreads();

  // Prime the TDM pipeline with the s=1 tile.
  if (tid < 32) {
    tdm_load_tile(yp + ((size_t)b0 * SEQ + 1) * TAG,
                  (uint32_t)(uintptr_t)&Ybuf[1][0][0]);
  }

  const int r = tid >> 4, j = tid & 15;  // 16 threads per batch row

  for (int s = 1; s < SEQ; ++s) {
    // Wave 0: issue next tile, wait for current. Wave 1: stage mask values.
    if (tid < 32) {
      if (s + 1 < SEQ) {
        tdm_load_tile(yp + ((size_t)b0 * SEQ + (s + 1)) * TAG,
                      (uint32_t)(uintptr_t)&Ybuf[(s + 1) & 1][0][0]);
        __builtin_amdgcn_s_wait_tensorcnt((short)1);
      } else {
        __builtin_amdgcn_s_wait_tensorcnt((short)0);
      }
    } else if (tid < 32 + BG) {
      int br = tid - 32;
      Msk[br] = mk[(size_t)(b0 + br) * SEQ + s];
    }
    __syncthreads();

    // Row max of PRE (wave32 shfl tree over 16-lane groups).
    float mx = -3.0e38f;
#pragma unroll
    for (int k = 0; k < 8; ++k) mx = fmaxf(mx, PREsh[r][j + 16 * k]);
#pragma unroll
    for (int off = 8; off > 0; off >>= 1) mx = fmaxf(mx, __shfl_xor(mx, off, 16));
    if (j == 0) Mrow[r] = mx;
    __syncthreads();

    // P = exp(PRE - rowmax)
#pragma unroll
    for (int k = 0; k < 8; ++k) {
      int idx = tid + 256 * k;
      int rr = idx >> 7, cc = idx & 127;
      Psh[rr][cc] = __expf(PREsh[rr][cc] - Mrow[rr]);
    }
    __syncthreads();

    // V = P x exp(A): 32 chained WMMA f32 16x16x4, two accumulators.
    v8f c0 = {0.f, 0.f, 0.f, 0.f, 0.f, 0.f, 0.f, 0.f};
    v8f c1 = {0.f, 0.f, 0.f, 0.f, 0.f, 0.f, 0.f, 0.f};
#pragma unroll
    for (int k = 0; k < 32; k += 2) {
      v2f a0;
      a0.x = Psh[l16][4 * k + 2 * kh];
      a0.y = Psh[l16][4 * k + 2 * kh + 1];
      c0 = __builtin_amdgcn_wmma_f32_16x16x4_f32(false, a0, false, bf[k],
                                                 (short)0, c0, false, false);
      v2f a1;
      a1.x = Psh[l16][4 * (k + 1) + 2 * kh];
      a1.y = Psh[l16][4 * (k + 1) + 2 * kh + 1];
      c1 = __builtin_amdgcn_wmma_f32_16x16x4_f32(false, a1, false, bf[k + 1],
                                                 (short)0, c1, false, false);
    }

    // PRE' = m*(rowmax + log(V) + y_pred[s]) + (1-m)*PRE
    const float* ybrow = &Ybuf[s & 1][0][0];
#pragma unroll
    for (int i = 0; i < 8; ++i) {
      int   br   = (lane < 16) ? i : (i + 8);
      int   u    = u0 + l16;
      float v    = c0[i] + c1[i];
      float logz = Mrow[br] + __logf(v) + ybrow[br * TAG + u];
      float mm   = Msk[br];
      PREsh[br][u] = mm * logz + (1.0f - mm) * PREsh[br][u];
    }
    __syncthreads();
  }

  // Final: log_Z[b] = logsumexp_u PRE[b,:]; accumulate mean loss.
  float mx = -3.0e38f;
#pragma unroll
  for (int k = 0; k < 8; ++k) mx = fmaxf(mx, PREsh[r][j + 16 * k]);
#pragma unroll
  for (int off = 8; off > 0; off >>= 1) mx = fmaxf(mx, __shfl_xor(mx, off, 16));
  float sm = 0.0f;
#pragma unroll
  for (int k = 0; k < 8; ++k) sm += __expf(PREsh[r][j + 16 * k] - mx);
#pragma unroll
  for (int off = 8; off > 0; off >>= 1) sm += __shfl_xor(sm, off, 16);
  if (j == 0) {
    float logZ = mx + __logf(sm);
    float loss = -(score[b0 + r] - logZ);
    atomicAdd(out, loss * (1.0f / BATCH));
  }
}

// ---------------------------------------------------------------------------
extern "C" void kernel_launch(void* const* d_in, const int* in_sizes, int n_in,
                              void* d_out, int out_size, void* d_ws, size_t ws_size,
                              hipStream_t stream) {
  (void)in_sizes; (void)n_in; (void)out_size; (void)ws_size;
  const float* yp = (const float*)d_in[0];
  const int*   yt = (const int*)d_in[1];
  const float* mk = (const float*)d_in[2];
  const float* A  = (const float*)d_in[3];
  float* out   = (float*)d_out;
  float* score = (float*)d_ws;  // 128 floats

  crf_zero_kernel<<<1, 1, 0, stream>>>(out);
  crf_score_kernel<<<BATCH, 256, 0, stream>>>(yp, yt, mk, A, score);
  crf_scan_kernel<<<NBLK, 256, 0, stream>>>(yp, mk, A, score, out);
}